// bgridSplatting2DTo3D_19164144075645
// MI455X (gfx1250) — compile-verified
//
#include <hip/hip_runtime.h>

// Bilateral-grid splat, degenerate-exact form:
// fl(fl(i/511)*511) == i exactly for all i in [0,511] (half-ulp analysis),
// so th = tw = 0 exactly and the trilinear scatter collapses to a gather:
//   out[n,c,i,j,r] = v*(1-tr) at r=ir0, v*tr at r=ir0+1, +0 elsewhere,
// with pr = gm*15 (< 15 always for gm in [0,1)), ir0 = floor(pr), tr = frac.
//
// Roofline: 302 MB streamed writes + ~21 MB reads => ~14 us at 23.3 TB/s.
// No GEMM structure: WMMA/SWMMAC not applicable. Optimization target is
// VALU-per-byte: one thread per pixel-channel (64 B of output), wave-
// cooperative transposed zero-fill (4 coalesced b128 stores per wave cover
// the wave's 2 KB), then 2 scalar dword stores per lane write the nonzero
// cells. Same-wave store ordering (stores in order with stores) guarantees
// the scatter lands after the fill; L2 merges the sectors.

typedef float float4v __attribute__((ext_vector_type(4)));

#define SR_ 16

__global__ __launch_bounds__(256) void bgrid_splat_kernel(
    const float* __restrict__ x,    // (2, 8, 512, 512)
    const float* __restrict__ gm,   // (2, 1, 512, 512)
    float* __restrict__ out)        // bg (2,8,512,512,16) ++ on (2,1,512,512,16)
{
    // one thread per pixel-channel: 18 slabs * 2^18 pixels = 4,718,592 threads
    const unsigned tid    = blockIdx.x * 256u + threadIdx.x;
    const unsigned lane   = threadIdx.x & 31u;
    const unsigned slab   = tid >> 18;       // 0..15 bg (n,c), 16..17 ones (n)
    const unsigned within = tid & 0x3FFFFu;  // i*512 + j

    // ---- wave-cooperative zero fill of the wave's 2 KB output block ----
    // store k: lane writes waveBase + k*512B + lane*16B  (512 B contiguous
    // per b128 store instruction across the wave)
    const float4v z = {0.0f, 0.0f, 0.0f, 0.0f};
    float4v* waveQuad = (float4v*)(out + ((size_t)(tid & ~31u) << 4)) + lane;
    waveQuad[0]  = z;
    waveQuad[32] = z;
    waveQuad[64] = z;
    waveQuad[96] = z;

    // ---- per-pixel-channel value + guide ----
    float v = 1.0f;
    unsigned n;
    if (slab < 16u) {              // wave-uniform branch (slab constant per wave)
        v = x[tid];                // x flat layout == tid for slabs 0..15
        n = slab >> 3;
    } else {
        n = slab - 16u;
    }
    const float g = gm[(n << 18) | within];

    // ---- r-dimension linear weights (same op order as reference) ----
    const float pr = g * (float)(SR_ - 1);
    const float f0 = floorf(pr);
    const float tr = pr - f0;
    int ir0 = (int)f0;
    ir0 = (ir0 < 0) ? 0 : ((ir0 > SR_ - 1) ? SR_ - 1 : ir0);
    const int ir1 = (ir0 + 1 > SR_ - 1) ? SR_ - 1 : ir0 + 1;

    const float c0 = v * (1.0f - tr);
    float c1 = v * tr;
    if (ir1 == ir0) c1 = c0 + c1;  // unreachable for gm in [0,1); keeps the
                                   // reference's two-add semantics if clamped

    // ---- scatter the two nonzero cells into this lane's own 64 B ----
    float* my = out + ((size_t)tid << 4);
    my[ir0] = c0;
    my[ir1] = c1;  // ordered after my[ir0]; overwrites with c0+c1 on collision
}

extern "C" void kernel_launch(void* const* d_in, const int* in_sizes, int n_in,
                              void* d_out, int out_size, void* d_ws, size_t ws_size,
                              hipStream_t stream) {
    const float* x  = (const float*)d_in[0];   // (2,8,512,512) float32
    const float* gm = (const float*)d_in[1];   // (2,1,512,512) float32
    float* out = (float*)d_out;                // 75,497,472 floats

    const unsigned totalThreads = 18u << 18;   // 4,718,592
    const unsigned block = 256;                // 8 wave32 waves
    const unsigned grid  = totalThreads / block; // 18,432 (exact)

    bgrid_splat_kernel<<<grid, block, 0, stream>>>(x, gm, out);
}